// SineGen_6021544149514
// MI455X (gfx1250) — compile-verified
//
#include <hip/hip_runtime.h>
#include <math.h>
#include <stdint.h>

// ---------------------------------------------------------------------------
// SineGen for MI455X (gfx1250, wave32).
// Math: %1.0 pre-cumsum and wrap correction are no-ops under sin(); the whole
// problem reduces to ONE prefix sum of f0 per batch, carried mod 48000 for
// fp32 precision. Bandwidth-bound: ~85MB traffic -> ~3.6us at 23.3 TB/s.
// Scan hierarchy: chunk sums -> serial chunk scan -> WMMA matmul-scan per
// 256-sample tile (V_WMMA_F32_16X16X4_F32, co-executes with VALU sin/noise).
// f0 chunks staged to LDS via global_load_async_to_lds_b32 (ASYNCcnt path).
// ---------------------------------------------------------------------------

typedef float v2f __attribute__((ext_vector_type(2)));
typedef float v8f __attribute__((ext_vector_type(8)));

#define T_LEN        480000
#define BATCH        4
#define DIM          9
#define TILE         256
#define TILES_PER_CHUNK 15
#define CHUNK        (TILE * TILES_PER_CHUNK)     // 3840
#define NCHUNK       (T_LEN / CHUNK)              // 125
#define SR_F         48000.0f
#define INV_SR       (1.0f / 48000.0f)
#define SINE_AMP     0.1f
#define NOISE_STD    0.003f
#define TWO_PI       6.283185307179586f

__device__ __forceinline__ uint32_t pcg(uint32_t x) {
    x = x * 747796405u + 2891336453u;
    x = ((x >> ((x >> 28u) + 4u)) ^ x) * 277803737u;
    return (x >> 22u) ^ x;
}

// cheap bounded mod-48000 (inputs < ~2e6): one floor, no libm fmodf loop
__device__ __forceinline__ float reduce48k(float x) {
    return x - SR_F * floorf(x * INV_SR);
}

// ---------------------------------------------------------------------------
// Kernel A: per-chunk sums of f0 (mod 48000).  grid=(125,4), block=256.
// ---------------------------------------------------------------------------
__global__ void __launch_bounds__(256)
chunk_sum_kernel(const float* __restrict__ f0, float* __restrict__ sums) {
    const int b = blockIdx.y, c = blockIdx.x, tid = threadIdx.x;
    const float* p = f0 + (size_t)b * T_LEN + (size_t)c * CHUNK;
    float s = 0.0f;
#pragma unroll
    for (int j = 0; j < TILES_PER_CHUNK; ++j) s += p[j * 256 + tid];
    for (int off = 16; off > 0; off >>= 1) s += __shfl_down(s, off, 32);
    __shared__ float wsum[8];
    if ((tid & 31) == 0) wsum[tid >> 5] = s;
    __syncthreads();
    if (tid == 0) {
        float t = 0.0f;
#pragma unroll
        for (int w = 0; w < 8; ++w) t += wsum[w];
        sums[b * NCHUNK + c] = reduce48k(t);
    }
}

// ---------------------------------------------------------------------------
// Kernel B: exclusive scan of 125 chunk sums per batch (tiny, serial).
// ---------------------------------------------------------------------------
__global__ void chunk_scan_kernel(const float* __restrict__ sums,
                                  float* __restrict__ offs) {
    const int b = threadIdx.x;
    if (b < BATCH) {
        float run = 0.0f;
        for (int i = 0; i < NCHUNK; ++i) {
            offs[b * NCHUNK + i] = run;
            run = reduce48k(run + sums[b * NCHUNK + i]);
        }
    }
}

// ---------------------------------------------------------------------------
// Kernel C: async-load chunk to LDS; WMMA matmul-scan per 256-sample tile;
// serial tile-offset scan; fused sine + noise + b128 stores.
//
// Tile: sample s = col*16 + k (column-major 16x16 matrix X).
//   P1 = L_incl * X         column-wise inclusive prefix
//   P  = P1 + CT * U_excl   add exclusive cross-column offsets
// Each 16x16x16 product = 4 chained V_WMMA_F32_16X16X4_F32 (K=4 slices).
// ---------------------------------------------------------------------------
__global__ void __launch_bounds__(256)
sine_gen_kernel(const float* __restrict__ f0, const float* __restrict__ chunkOff,
                float* __restrict__ out) {
    __shared__ float ldsF0[CHUNK];
    __shared__ float ldsP[CHUNK];
    __shared__ float ldsTot[TILES_PER_CHUNK];
    __shared__ float ldsOff[TILES_PER_CHUNK];

    const int b = blockIdx.y, c = blockIdx.x, tid = threadIdx.x;
    const size_t chunkStart = (size_t)c * CHUNK;
    const float* gsrc = f0 + (size_t)b * T_LEN + chunkStart;

    // ---- async copy f0 chunk -> LDS (gfx1250 async-to-LDS path) ----
    {
        uint32_t lbase = (uint32_t)(uintptr_t)&ldsF0[tid];   // low 32b = LDS offset
        uint64_t gbase = (uint64_t)(uintptr_t)&gsrc[tid];
#pragma unroll
        for (int j = 0; j < TILES_PER_CHUNK; ++j) {
            uint32_t laddr = lbase + (uint32_t)(j * 1024);
            uint64_t gaddr = gbase + (uint64_t)(j * 1024);
            asm volatile("global_load_async_to_lds_b32 %0, %1, off"
                         :: "v"(laddr), "v"(gaddr) : "memory");
        }
        asm volatile("s_wait_asynccnt 0" ::: "memory");
    }
    __syncthreads();

    const int lane = tid & 31, wave = tid >> 5;
    const int col = lane & 15, hi = lane >> 4;

    // ---- phase 1: WMMA prefix scan per tile ----
    for (int t = wave; t < TILES_PER_CHUNK; t += 8) {
        const int tb = t * TILE;
        v2f xb[4], la[4], ub[4];
#pragma unroll
        for (int i = 0; i < 4; ++i) {
            const int k0 = 4 * i + 2 * hi;
            xb[i] = *(const v2f*)&ldsF0[tb + col * 16 + k0];   // X in B-layout
            la[i][0] = (col >= k0)     ? 1.0f : 0.0f;          // L_incl in A-layout
            la[i][1] = (col >= k0 + 1) ? 1.0f : 0.0f;
            ub[i][0] = (k0     < col)  ? 1.0f : 0.0f;          // U_excl in B-layout
            ub[i][1] = (k0 + 1 < col)  ? 1.0f : 0.0f;
        }
        v8f acc = {};
#pragma unroll
        for (int i = 0; i < 4; ++i)
            acc = __builtin_amdgcn_wmma_f32_16x16x4_f32(
                false, la[i], false, xb[i], (short)0, acc, false, false);

        const float myct = acc[7];                 // P1[15,c] in lanes 16..31
        v2f cta[4];
#pragma unroll
        for (int i = 0; i < 4; ++i) {
            const int k0 = 4 * i + 2 * hi;
            cta[i][0] = __shfl(myct, 16 + k0, 32);
            cta[i][1] = __shfl(myct, 16 + k0 + 1, 32);
        }
#pragma unroll
        for (int i = 0; i < 4; ++i)
            acc = __builtin_amdgcn_wmma_f32_16x16x4_f32(
                false, cta[i], false, ub[i], (short)0, acc, false, false);

        float4* dst = (float4*)&ldsP[tb + col * 16 + 8 * hi];  // 8 consecutive samples
        dst[0] = make_float4(acc[0], acc[1], acc[2], acc[3]);
        dst[1] = make_float4(acc[4], acc[5], acc[6], acc[7]);
        if (lane == 31) ldsTot[t] = acc[7];                    // tile total
    }
    __syncthreads();

    // ---- tile-offset scan (15 values, seeded with chunk offset) ----
    if (tid == 0) {
        float run = chunkOff[b * NCHUNK + c];
#pragma unroll
        for (int t = 0; t < TILES_PER_CHUNK; ++t) {
            ldsOff[t] = run;
            run = reduce48k(run + ldsTot[t]);
        }
    }
    __syncthreads();

    // ---- phase 2: fused sine + noise; 4 samples/thread -> 9x b128 stores ----
    float phi[DIM];
#pragma unroll
    for (int k = 0; k < DIM; ++k) {
        uint32_t h = pcg(0xA5E1u + (uint32_t)(b * DIM + k));
        phi[k] = (k == 0) ? 0.0f : (float)(h >> 8) * (1.0f / 16777216.0f);
    }

#pragma unroll 1
    for (int j = 0; j < 4; ++j) {                  // rounds of 1024 samples
        const int s4 = j * 1024 + tid * 4;
        if (s4 < CHUNK) {
            const float toff = ldsOff[s4 >> 8];    // 4 samples never cross a tile
            const float4 pv  = *(const float4*)&ldsP[s4];
            const float4 fv  = *(const float4*)&ldsF0[s4];
            const size_t tglob = chunkStart + (size_t)s4;
            float* op = out + ((size_t)b * T_LEN + tglob) * DIM;
            const uint32_t nbase =
                ((uint32_t)b * (uint32_t)T_LEN + (uint32_t)tglob) * DIM;

            float buf[4 * DIM];
            const float Sv[4] = {toff + pv.x, toff + pv.y, toff + pv.z, toff + pv.w};
            const float f0v[4] = {fv.x, fv.y, fv.z, fv.w};
#pragma unroll
            for (int i = 0; i < 4; ++i) {
                const float uv = (f0v[i] > 0.0f) ? 1.0f : 0.0f;
                const float namp = uv * NOISE_STD + (1.0f - uv) * (SINE_AMP / 3.0f);
                const uint32_t nb = nbase + (uint32_t)(i * DIM);
#pragma unroll
                for (int k = 0; k < DIM; ++k) {
                    // Sv < ~150400 -> arg < 29: fp32 frac precision ~2e-6 cycles, OK
                    float arg = (float)(k + 1) * Sv[i] * INV_SR + phi[k];
                    arg -= floorf(arg);
                    const float sw = SINE_AMP * __sinf(TWO_PI * arg);
                    const uint32_t h1 = pcg(nb + (uint32_t)k);
                    const uint32_t h2 = pcg((nb + (uint32_t)k) ^ 0x9E3779B9u);
                    const float u1 = (float)(h1 >> 8) * (1.0f / 16777216.0f) + 5.96e-8f;
                    const float u2 = (float)(h2 >> 8) * (1.0f / 16777216.0f);
                    const float nrm = __fsqrt_rn(-2.0f * __logf(u1)) * __cosf(TWO_PI * u2);
                    buf[i * DIM + k] = sw * uv + namp * nrm;
                }
            }
            // 36 floats, 144B-aligned & contiguous -> 9 global_store_b128
            float4* op4 = (float4*)op;
#pragma unroll
            for (int q = 0; q < DIM; ++q)
                op4[q] = make_float4(buf[4 * q], buf[4 * q + 1],
                                     buf[4 * q + 2], buf[4 * q + 3]);
        }
    }
}

// ---------------------------------------------------------------------------
extern "C" void kernel_launch(void* const* d_in, const int* in_sizes, int n_in,
                              void* d_out, int out_size, void* d_ws, size_t ws_size,
                              hipStream_t stream) {
    const float* f0 = (const float*)d_in[0];
    float* out  = (float*)d_out;
    float* sums = (float*)d_ws;                 // [BATCH*NCHUNK]
    float* offs = sums + BATCH * NCHUNK;        // [BATCH*NCHUNK]

    dim3 grid(NCHUNK, BATCH);
    chunk_sum_kernel<<<grid, 256, 0, stream>>>(f0, sums);
    chunk_scan_kernel<<<1, 32, 0, stream>>>(sums, offs);
    sine_gen_kernel<<<grid, 256, 0, stream>>>(f0, offs, out);
}